// UnSqueezeConsLayer_61744449847330
// MI455X (gfx1250) — compile-verified
//
#include <hip/hip_runtime.h>
#include <stdint.h>

// Problem shape (fixed by the reference).
#define BB 16
#define HH 1024
#define WW 1024

typedef float f32x4 __attribute__((ext_vector_type(4)));
typedef int   i32x4 __attribute__((ext_vector_type(4)));
typedef __attribute__((address_space(1))) i32x4 as1_i32x4;  // global int4
typedef __attribute__((address_space(3))) i32x4 as3_i32x4;  // LDS int4

// ---- CDNA5 async global->LDS copy (128b per lane, tracked by ASYNCcnt) ----
#if defined(__HIP_DEVICE_COMPILE__) && __has_builtin(__builtin_amdgcn_global_load_async_to_lds_b128)
#define ATHENA_HAVE_ASYNC_B128 1
#endif
#if defined(__HIP_DEVICE_COMPILE__) && __has_builtin(__builtin_amdgcn_s_wait_asynccnt)
#define ATHENA_HAVE_WAIT_ASYNC 1
#endif

__device__ __forceinline__ void async_g2l_b128(const void* g, void* lds) {
#ifdef ATHENA_HAVE_ASYNC_B128
  __builtin_amdgcn_global_load_async_to_lds_b128(
      (as1_i32x4*)(void*)g, (as3_i32x4*)lds, /*offset=*/0, /*cpol=*/0);
#else
  // vdst = LDS byte offset (32-bit VGPR), vaddr = 64-bit global address, saddr = off
  uint32_t lds_off = (uint32_t)(size_t)(__attribute__((address_space(3))) void*)lds;
  asm volatile("global_load_async_to_lds_b128 %0, %1, off"
               :
               : "v"(lds_off), "v"((uint64_t)(size_t)g)
               : "memory");
#endif
}

__device__ __forceinline__ void wait_async0() {
#ifdef ATHENA_HAVE_WAIT_ASYNC
  __builtin_amdgcn_s_wait_asynccnt(0);
  asm volatile("" ::: "memory");  // compiler ordering fence for the LDS reads
#else
  asm volatile("s_wait_asynccnt 0x0" ::: "memory");
#endif
}

// One block handles one (b, r): 256 lanes x 4 columns = W.
// Produces output rows 2r and 2r+1 of both x and cons for that batch.
__global__ __launch_bounds__(256) void unsqueeze_cons_kernel(
    const float* __restrict__ in, float* __restrict__ out_x,
    float* __restrict__ out_c) {
  const uint32_t t = threadIdx.x;              // 0..255  == column block
  const uint32_t r = blockIdx.x & (HH - 1);    // row within image
  const uint32_t b = blockIdx.x >> 10;         // batch (HH == 1024)
  const uint32_t c = t << 2;                   // first of 4 input columns

  const size_t chs = (size_t)HH * WW;          // channel stride (elements)
  const float* pb = in + ((size_t)b * 4) * chs + (size_t)r * WW + c;  // this batch
  const float* p0 = in + (size_t)r * WW + c;                          // batch 0, ch 0

  // ---- kick off async staging of cons sources (batch-0 ch0..2, L2-hot) ----
  __shared__ f32x4 stage[3][256];
  async_g2l_b128(p0,           &stage[0][t]);
  async_g2l_b128(p0 + chs,     &stage[1][t]);
  async_g2l_b128(p0 + 2 * chs, &stage[2][t]);

  // ---- x path: 4x B128 loads, register interleave, 4x B128 NT stores ----
  f32x4 q0 = *reinterpret_cast<const f32x4*>(pb);            // ch0 -> (2r,   even)
  f32x4 q1 = *reinterpret_cast<const f32x4*>(pb + chs);      // ch1 -> (2r+1, even)
  f32x4 q2 = *reinterpret_cast<const f32x4*>(pb + 2 * chs);  // ch2 -> (2r,   odd)
  f32x4 q3 = *reinterpret_cast<const f32x4*>(pb + 3 * chs);  // ch3 -> (2r+1, odd)

  f32x4 o00 = {q0.x, q2.x, q0.y, q2.y};
  f32x4 o01 = {q0.z, q2.z, q0.w, q2.w};
  f32x4 o10 = {q1.x, q3.x, q1.y, q3.y};
  f32x4 o11 = {q1.z, q3.z, q1.w, q3.w};

  const size_t W2 = 2 * (size_t)WW;
  const size_t row0 = ((size_t)b * (2 * HH) + 2 * (size_t)r) * W2 + 2 * (size_t)c;
  const size_t row1 = row0 + W2;

  __builtin_nontemporal_store(o00, reinterpret_cast<f32x4*>(out_x + row0));
  __builtin_nontemporal_store(o01, reinterpret_cast<f32x4*>(out_x + row0 + 4));
  __builtin_nontemporal_store(o10, reinterpret_cast<f32x4*>(out_x + row1));
  __builtin_nontemporal_store(o11, reinterpret_cast<f32x4*>(out_x + row1 + 4));

  // ---- cons path: consume staged LDS data ----
  wait_async0();
  f32x4 a0 = stage[0][t];
  f32x4 a1 = stage[1][t];
  f32x4 a2 = stage[2][t];

  f32x4 s2 = (a0 + 1.0f) / 2.0f;            // -> (2r+1, even)
  f32x4 s1 = (a0 + a1 + 1.0f) / 3.0f;       // -> (2r,   odd)
  f32x4 s0 = (a0 + a1 + a2 + 1.0f) / 4.0f;  // -> (2r+1, odd)

  f32x4 c00 = {1.0f, s1.x, 1.0f, s1.y};
  f32x4 c01 = {1.0f, s1.z, 1.0f, s1.w};
  f32x4 c10 = {s2.x, s0.x, s2.y, s0.y};
  f32x4 c11 = {s2.z, s0.z, s2.w, s0.w};

  __builtin_nontemporal_store(c00, reinterpret_cast<f32x4*>(out_c + row0));
  __builtin_nontemporal_store(c01, reinterpret_cast<f32x4*>(out_c + row0 + 4));
  __builtin_nontemporal_store(c10, reinterpret_cast<f32x4*>(out_c + row1));
  __builtin_nontemporal_store(c11, reinterpret_cast<f32x4*>(out_c + row1 + 4));
}

extern "C" void kernel_launch(void* const* d_in, const int* in_sizes, int n_in,
                              void* d_out, int out_size, void* d_ws, size_t ws_size,
                              hipStream_t stream) {
  (void)in_sizes; (void)n_in; (void)out_size; (void)d_ws; (void)ws_size;
  const float* in = (const float*)d_in[0];
  float* out_x = (float*)d_out;                                  // x: (B,1,2H,2W)
  float* out_c = out_x + (size_t)BB * (2 * HH) * (2 * WW);       // cons: same size

  dim3 grid(BB * HH);   // 16384 blocks: one (batch,row) pair each
  dim3 block(256);      // 8 wave32 waves; 256 lanes x 4 cols = W
  unsqueeze_cons_kernel<<<grid, block, 0, stream>>>(in, out_x, out_c);
}